// ExponentialSmoothing_65824668779038
// MI455X (gfx1250) — compile-verified
//
#include <hip/hip_runtime.h>
#include <math.h>

// Holt-Winters triple exponential smoothing, B=8192 series x L=4096 steps, P=7.
// Bandwidth-bound (~512MB @ 23.3TB/s => ~22us floor). One lane per series.
// CDNA5 path: TDM tensor_load_to_lds (double-buffered, LDS row padding via D#),
// s_wait_tensorcnt, LDS-staged coalesced nontemporal float4 output stores.

#define B_SERIES 8192
#define L_STEPS  4096
#define CHUNK    32                 // timesteps per tile (L divisible by 32)
#define TILES    (L_STEPS / CHUNK)  // 128
#define ROWS     32                 // series per wave (== wave32)
#define IN_STRIDE  (CHUNK + 1)      // 33 dwords: TDM pads 1 dword per 32-dword row
#define OUT_STRIDE 100              // 96 data dwords + 4 pad (16B-aligned rows)

typedef unsigned int u32x4 __attribute__((ext_vector_type(4)));
typedef int          i32x4 __attribute__((ext_vector_type(4)));
typedef int          i32x8 __attribute__((ext_vector_type(8)));
typedef float        f32x4 __attribute__((ext_vector_type(4)));  // native vector for NT store

// Build a 2D D# (group0/group1) and issue TENSOR_LOAD_TO_LDS:
//   tile: CHUNK x ROWS f32, row stride = L_STEPS elements, 1-dword LDS pad per row.
static __device__ __forceinline__ void tdm_load_tile(unsigned lds_byte_addr,
                                                     const float* gsrc) {
  unsigned long long ga = (unsigned long long)(uintptr_t)gsrc;
  u32x4 g0;
  g0.x = 1u;                                   // count=1 (valid), no gather
  g0.y = lds_byte_addr;                        // lds_addr [63:32]
  g0.z = (unsigned)(ga & 0xFFFFFFFFull);       // global_addr[31:0]
  g0.w = (unsigned)((ga >> 32) & 0x1FFFFFFull) // global_addr[56:32]
       | (2u << 30);                           // type=2 ("image")
  i32x8 g1;
  g1[0] = (int)((2u << 16)                     // data_size = 4 bytes
              | (1u << 20)                     // pad_enable
              | (4u << 22)                     // pad_interval code 4 => every 32 dwords
              | (0u << 25));                   // pad_amount  code 0 => 1 dword
  g1[1] = (int)((CHUNK & 0xFFFFu) << 16);                              // tensor_dim0 lo
  g1[2] = (int)(((CHUNK >> 16) & 0xFFFFu) | ((ROWS & 0xFFFFu) << 16)); // tdim0 hi | tdim1 lo
  g1[3] = (int)(((ROWS >> 16) & 0xFFFFu) | ((CHUNK & 0xFFFFu) << 16)); // tdim1 hi | tile_dim0
  g1[4] = (int)(ROWS & 0xFFFFu);                                       // tile_dim1 (tile_dim2=0)
  g1[5] = (int)(L_STEPS);                                              // tensor_dim0_stride lo32
  g1[6] = 0;                                                           // stride hi | dim1stride lo
  g1[7] = 0;
#if __clang_major__ >= 23
  __builtin_amdgcn_tensor_load_to_lds(g0, g1, (i32x4)0, (i32x4)0, (i32x8)0, 0);
#else
  __builtin_amdgcn_tensor_load_to_lds(g0, g1, (i32x4)0, (i32x4)0, 0);
#endif
}

__global__ __launch_bounds__(32) void ExponentialSmoothing_65824668779038_kernel(
    const float* __restrict__ x,
    const float* __restrict__ alpha,
    const float* __restrict__ beta,
    const float* __restrict__ gamma,
    float* __restrict__ out) {
  __shared__ __align__(16) float lds_in[2][ROWS * IN_STRIDE];
  __shared__ __align__(16) float lds_out[ROWS * OUT_STRIDE];

  const int lane = threadIdx.x;            // 0..31, one series per lane
  const int b0   = blockIdx.x * ROWS;      // first series of this wave

  const float a  = 1.0f / (1.0f + __expf(-alpha[0]));
  const float bb = 1.0f / (1.0f + __expf(-beta[0]));
  const float g  = 1.0f / (1.0f + __expf(-gamma[0]));
  const float ia = 1.0f - a, ib = 1.0f - bb, ig = 1.0f - g;

  // Prime pipeline: tile 0 -> buffer 0.
  tdm_load_tile((unsigned)(uintptr_t)&lds_in[0][0], x + (size_t)b0 * L_STEPS);

  float level = 0.0f, trend = 0.0f;
  // Seasonal buffer as rotating registers: r0 is the element consumed next.
  float r0 = 0.f, r1 = 0.f, r2 = 0.f, r3 = 0.f, r4 = 0.f, r5 = 0.f, r6 = 0.f;

  for (int k = 0; k < TILES; ++k) {
    // Issue next tile, then wait for current (in-order per-wave TENSORcnt).
    if (k + 1 < TILES) {
      tdm_load_tile((unsigned)(uintptr_t)&lds_in[(k + 1) & 1][0],
                    x + (size_t)b0 * L_STEPS + (size_t)(k + 1) * CHUNK);
      __builtin_amdgcn_s_wait_tensorcnt(1);   // tile k complete
    } else {
      __builtin_amdgcn_s_wait_tensorcnt(0);
    }
    __asm__ volatile("" ::: "memory");        // compiler fence around DMA-written LDS

    const float* in   = &lds_in[k & 1][lane * IN_STRIDE];  // stride 33: conflict-free
    float*       orow = &lds_out[lane * OUT_STRIDE];

    int i0 = 0;
    if (k == 0) {
      // level0 = x[0]; trend0 = x[1]-x[0]; buf[j] = x[j]-level0; seas0 = buf[0]
      float x0 = in[0], x1 = in[1];
      level = x0;
      trend = x1 - x0;
      float s0 = in[0] - x0, s1 = in[1] - x0, s2 = in[2] - x0, s3 = in[3] - x0,
            s4 = in[4] - x0, s5 = in[5] - x0, s6 = in[6] - x0;
      // First consumed index is ti = 1 (t=1), so rotate: [b1..b6, b0].
      r0 = s1; r1 = s2; r2 = s3; r3 = s4; r4 = s5; r5 = s6; r6 = s0;
      orow[0] = level; orow[1] = trend; orow[2] = s0;   // outputs at t=0
      i0 = 1;
    }

#pragma unroll 8
    for (int i = i0; i < CHUNK; ++i) {
      float xt   = in[i];
      float prev = r0;
      float nl = a * (xt - prev) + ia * (level + trend);
      float nt = bb * (nl - level) + ib * trend;
      float ns = g * (xt - nl) + ig * prev;
      r0 = r1; r1 = r2; r2 = r3; r3 = r4; r4 = r5; r5 = r6; r6 = ns;  // elided by unroll
      level = nl; trend = nt;
      orow[3 * i + 0] = nl;
      orow[3 * i + 1] = nt;
      orow[3 * i + 2] = ns;
    }
    __asm__ volatile("" ::: "memory");

    // Coalesced drain: 32 rows x 96 dwords = 768 float4; each series-tile's
    // 96 output floats are contiguous in the (B, L, 3) layout.
    f32x4* out4 = (f32x4*)out;
#pragma unroll 4
    for (int it = 0; it < 24; ++it) {
      int idx = it * 32 + lane;
      int r = idx / 24;               // source row (series)
      int q = idx - r * 24;           // float4 index within row
      f32x4 v = *(const f32x4*)&lds_out[r * OUT_STRIDE + 4 * q];
      size_t gq = (size_t)(((size_t)(b0 + r) * L_STEPS + (size_t)k * CHUNK) * 3 / 4) + q;
      __builtin_nontemporal_store(v, &out4[gq]);
    }
  }
}

extern "C" void kernel_launch(void* const* d_in, const int* in_sizes, int n_in,
                              void* d_out, int out_size, void* d_ws, size_t ws_size,
                              hipStream_t stream) {
  const float* x     = (const float*)d_in[0];
  const float* alpha = (const float*)d_in[1];
  const float* beta  = (const float*)d_in[2];
  const float* gamma = (const float*)d_in[3];
  float* out = (float*)d_out;
  dim3 grid(B_SERIES / ROWS);   // 256 one-wave workgroups -> spread across WGPs
  dim3 block(32);               // wave32: one wave per block, no barriers needed
  hipLaunchKernelGGL(ExponentialSmoothing_65824668779038_kernel, grid, block, 0, stream,
                     x, alpha, beta, gamma, out);
}